// ParallelCausalConv1d_36240934043693
// MI455X (gfx1250) — compile-verified
//
#include <hip/hip_runtime.h>
#include <cstdint>
#include <cstddef>

// Depthwise causal conv1d, B=4 S=8192 C=2048 K=4, fp32.
// Memory-bound (AI ~1 FLOP/B): floor ~23us at 23.3 TB/s. Strategy: stream x
// through LDS with CDNA5 async global->LDS B128 copies (ASYNCcnt), reuse each
// element 4x from LDS via a register sliding window, store coalesced B128
// with NT temporal hint (write-once output must not thrash L2).

constexpr int Bc = 4;
constexpr int Sc = 8192;
constexpr int Cc = 2048;
constexpr int Kc = 4;

constexpr int THREADS  = 256;
constexpr int S_TILE   = 64;            // output rows per block
constexpr int CH_TILE  = 256;           // channels per block
constexpr int HALO     = Kc - 1;        // 3
constexpr int ROWS     = S_TILE + HALO; // 67 input rows in LDS
constexpr int CF4      = CH_TILE / 4;   // 64 float4 columns
constexpr int ROWS_PER_GROUP = S_TILE / (THREADS / CF4); // 64 / 4 = 16

typedef int   v4i __attribute__((ext_vector_type(4)));
typedef float v4f __attribute__((ext_vector_type(4)));
typedef __attribute__((address_space(1))) v4i* global_v4i_ptr;
typedef __attribute__((address_space(3))) v4i* lds_v4i_ptr;

__device__ __forceinline__ void async_copy_f4(const float* gsrc, float* lds_dst) {
#if __has_builtin(__builtin_amdgcn_global_load_async_to_lds_b128)
  // CDNA5 GLOBAL_LOAD_ASYNC_TO_LDS_B128: per-lane 16B global->LDS, ASYNCcnt.
  __builtin_amdgcn_global_load_async_to_lds_b128(
      (global_v4i_ptr)(const_cast<float*>(gsrc)),
      (lds_v4i_ptr)(lds_dst),
      /*imm offset*/ 0, /*cpol*/ 0);
#else
  *(v4f*)lds_dst = *(const v4f*)gsrc;  // fallback: sync copy (still B128)
#endif
}

__device__ __forceinline__ void wait_async() {
#if __has_builtin(__builtin_amdgcn_s_wait_asynccnt)
  __builtin_amdgcn_s_wait_asynccnt(0);
#else
  asm volatile("s_wait_asynccnt 0x0" ::: "memory");
#endif
}

__global__ __launch_bounds__(THREADS)
void ParallelCausalConv1d_36240934043693_kernel(const float* __restrict__ x,
                                                const float* __restrict__ w,
                                                const float* __restrict__ bias,
                                                float* __restrict__ out) {
  __shared__ __align__(16) float tile[ROWS * CH_TILE];  // 67 KB

  const int tid = threadIdx.x;
  const int s0  = blockIdx.x * S_TILE;
  const int c0  = blockIdx.y * CH_TILE;
  const int b   = blockIdx.z;

  const float* xb = x + (size_t)b * Sc * Cc;

  // ---- Load phase: 67 rows x 64 float4 -> LDS via async B128 copies ----
  constexpr int NF4 = ROWS * CF4;  // 4288
  for (int i = tid; i < NF4; i += THREADS) {
    const int row  = i >> 6;          // i / CF4
    const int cv   = i & (CF4 - 1);   // i % CF4
    const int s_in = s0 - HALO + row;
    float* ldst = &tile[row * CH_TILE + cv * 4];
    if (s_in >= 0) {
      async_copy_f4(xb + (size_t)s_in * Cc + c0 + cv * 4, ldst);
    } else {
      *(v4f*)ldst = (v4f)0.0f;        // causal left padding
    }
  }

  // ---- Weight/bias loads issued now: latency hides under the async DMA ----
  const int cv = tid & (CF4 - 1);
  const int r0 = (tid >> 6) * ROWS_PER_GROUP;  // group of 16 rows
  const int c  = c0 + cv * 4;

  const v4f w0 = *(const v4f*)(w + 0 * Cc + c);
  const v4f w1 = *(const v4f*)(w + 1 * Cc + c);
  const v4f w2 = *(const v4f*)(w + 2 * Cc + c);
  const v4f w3 = *(const v4f*)(w + 3 * Cc + c);
  const v4f bi = *(const v4f*)(bias + c);

  wait_async();
  __syncthreads();

  // ---- Compute phase: each thread = one float4 column x 16 output rows ----
  const float* col = &tile[cv * 4];
  // Sliding window over LDS rows r0 .. r0+18 (ds_load_b128).
  v4f xa = *(const v4f*)(col + (r0 + 0) * CH_TILE);
  v4f xv = *(const v4f*)(col + (r0 + 1) * CH_TILE);
  v4f xc = *(const v4f*)(col + (r0 + 2) * CH_TILE);

  float* op = out + ((size_t)b * Sc + s0 + r0) * Cc + c;
#pragma unroll
  for (int r = 0; r < ROWS_PER_GROUP; ++r) {
    const v4f xd = *(const v4f*)(col + (r0 + r + 3) * CH_TILE);
    v4f o;
    o.x = fmaf(w3.x, xd.x, fmaf(w2.x, xc.x, fmaf(w1.x, xv.x, fmaf(w0.x, xa.x, bi.x))));
    o.y = fmaf(w3.y, xd.y, fmaf(w2.y, xc.y, fmaf(w1.y, xv.y, fmaf(w0.y, xa.y, bi.y))));
    o.z = fmaf(w3.z, xd.z, fmaf(w2.z, xc.z, fmaf(w1.z, xv.z, fmaf(w0.z, xa.z, bi.z))));
    o.w = fmaf(w3.w, xd.w, fmaf(w2.w, xc.w, fmaf(w1.w, xv.w, fmaf(w0.w, xa.w, bi.w))));
    __builtin_nontemporal_store(o, (v4f*)op);  // global_store_b128, TH=NT
    op += Cc;
    xa = xv; xv = xc; xc = xd;
  }
}

extern "C" void kernel_launch(void* const* d_in, const int* in_sizes, int n_in,
                              void* d_out, int out_size, void* d_ws, size_t ws_size,
                              hipStream_t stream) {
  (void)in_sizes; (void)n_in; (void)out_size; (void)d_ws; (void)ws_size;
  const float* x    = (const float*)d_in[0];   // (B, S, C)
  const float* w    = (const float*)d_in[1];   // (K, 1, C) -> w[k*C + c]
  const float* bias = (const float*)d_in[2];   // (C,)
  float* out        = (float*)d_out;           // (B, S, C)

  dim3 grid(Sc / S_TILE, Cc / CH_TILE, Bc);    // (128, 8, 4)
  hipLaunchKernelGGL(ParallelCausalConv1d_36240934043693_kernel,
                     grid, dim3(THREADS), 0, stream, x, w, bias, out);
}